// QCircuitThresholdModel_90211493085951
// MI455X (gfx1250) — compile-verified
//
#include <hip/hip_runtime.h>
#include <hip/hip_bf16.h>
#include <math.h>

typedef __attribute__((ext_vector_type(2))) float v2f;
typedef __attribute__((ext_vector_type(8))) float v8f;

#define NN   262144     // nodes
#define NE   2097152    // edges
#define NB   1024       // graphs
#define HD   64         // hidden
#define GFDI 22
#define NTHO 10

__device__ __forceinline__ float silu_f(float x) { return x / (1.f + __expf(-x)); }

// float atomic max via monotonic int/uint punning (init with -inf)
__device__ __forceinline__ void atomicMaxFloat(float* addr, float val) {
    if (val >= 0.f) atomicMax((int*)addr, __float_as_int(val));
    else            atomicMin((unsigned int*)addr, __float_as_uint(val));
}

// ---- output tile store: compile-time LDO -> one base addr + immediate offsets ----
template<int LDO>
__device__ __forceinline__ void store_tile(float* out, int rowbase, int colg,
                                           float bias, v8f acc, int lane) {
    const int mo = (lane >> 4) * 8;
    float* p = out + (size_t)(rowbase + mo) * LDO + colg;
#pragma unroll
    for (int j = 0; j < 8; ++j) p[j * LDO] = acc[j] + bias;
}

// ---- WMMA f32 16x16x4 K-loop, A from xs (LDS), B from wl (LDS, pair-interleaved) ----
template<int KTILES>
__device__ __forceinline__ v8f wmma_kloop(const float* abase, const float* wbase) {
    v8f acc = {0.f, 0.f, 0.f, 0.f, 0.f, 0.f, 0.f, 0.f};
#pragma unroll
    for (int kt = 0; kt < KTILES; ++kt) {
        const float2 av = *(const float2*)(abase + kt * 4);    // ds_load_b64, imm offset
        const float2 bv = *(const float2*)(wbase + kt * 256);  // ds_load_b64, imm offset
        v2f a; a.x = av.x; a.y = av.y;
        v2f b; b.x = bv.x; b.y = bv.y;
        acc = __builtin_amdgcn_wmma_f32_16x16x4_f32(false, a, false, b,
                                                    (short)0, acc, false, false);
    }
    return acc;
}

// ---- stage one [FIN x 64] weight matrix into pair-interleaved LDS (zero-pad to KROWS) ----
// wl layout: (kr>>1)*128 + 2*col + (kr&1); float4 global loads, direct global pointer.
template<int KROWS, int FIN>
__device__ __forceinline__ void stage_mat(const float* __restrict__ W, float* dst,
                                          int t, int nthreads) {
    for (int i = t; i < KROWS * 16; i += nthreads) {
        const int kr = i >> 4, c4 = (i & 15) * 4;
        float4 v;
        if (kr < FIN) v = *(const float4*)(W + kr * 64 + c4);
        else          v = make_float4(0.f, 0.f, 0.f, 0.f);
        float* d = dst + (kr >> 1) * 128 + 2 * c4 + (kr & 1);
        d[0] = v.x; d[2] = v.y; d[4] = v.z; d[6] = v.w;
    }
}

// ---- one projection phase: stage 2 matrices, each wave computes one 16x16 tile ----
// PHASE 0 -> mats {q,k} (both into qkv, cb = ml*64); PHASE 1 -> mats {v (qkv cb128), s (agg)}
template<int KROWS, int KTILES, int FIN, int PHASE>
__device__ __forceinline__ void conv_phase(const float* xs, int xstr, float* wl,
                                           const float* W0, const float* b0,
                                           const float* W1, const float* b1,
                                           float* qkv, float* agg, int rowbase) {
    const int tid  = threadIdx.x;
    const int wid  = tid >> 5;
    const int lane = tid & 31;
    const int lm   = lane & 15;
    __syncthreads();                                   // previous wl reads done
    if (tid < 128) stage_mat<KROWS, FIN>(W0, wl, tid, 128);
    else           stage_mat<KROWS, FIN>(W1, wl + KROWS * 64, tid - 128, 128);
    __syncthreads();
    const int ml  = wid >> 2;                          // 0..1 (wave-uniform)
    const int lt  = wid & 3;
    const int col = lt * 16 + lm;
    const float* abase = xs + lm * xstr + ((lane >> 4) << 1);
    const float* wbase = wl + ml * (KROWS * 64) + ((lane >> 4) << 7) + 2 * col;
    const v8f acc = wmma_kloop<KTILES>(abase, wbase);  // EXEC all-ones here
    if (PHASE == 0) {
        if (ml == 0) store_tile<192>(qkv, rowbase, col,      b0[col], acc, lane);
        else         store_tile<192>(qkv, rowbase, 64 + col, b1[col], acc, lane);
    } else {
        if (ml == 0) store_tile<192>(qkv, rowbase, 128 + col, b0[col], acc, lane);
        else         store_tile<64>(agg, rowbase, col,        b1[col], acc, lane);
    }
}

// ---------------- conv1: featurize (19-dim, padded to 20) + 4 projections ----------------
__global__ void k_gemm_conv1(const int* gti, const int* arity, const int* isdir,
                             const float* idxn, const float* emb,
                             const float* Wq, const float* bq, const float* Wk, const float* bk,
                             const float* Wv, const float* bv, const float* Ws, const float* bs,
                             float* qkv, float* agg) {
    __shared__ float xs[16 * 22];
    __shared__ float wl[2 * 20 * 64];
    const int rowbase = blockIdx.x * 16;
    for (int i = threadIdx.x; i < 16 * 20; i += blockDim.x) {
        const int r = i / 20, c = i - r * 20;
        const int node = rowbase + r;
        float v;
        if (c < 16)       v = emb[gti[node] * 16 + c];
        else if (c == 16) v = (float)arity[node];
        else if (c == 17) v = (float)isdir[node];
        else if (c == 18) v = idxn[node];
        else              v = 0.f;                     // pad col 19
        xs[r * 22 + c] = v;
    }
    conv_phase<20, 5, 19, 0>(xs, 22, wl, Wq, bq, Wk, bk, qkv, agg, rowbase);
    conv_phase<20, 5, 19, 1>(xs, 22, wl, Wv, bv, Ws, bs, qkv, agg, rowbase);
}

// ---------------- conv2: X[N x 64] -> 4 projections (skip-output may alias X) ----------------
__global__ void k_gemm_conv2(const float* X,
                             const float* Wq, const float* bq, const float* Wk, const float* bk,
                             const float* Wv, const float* bv, const float* Ws, const float* bs,
                             float* qkv, float* agg) {
    __shared__ float xs[16 * 68];
    __shared__ float wl[2 * 64 * 64];
    const int rowbase = blockIdx.x * 16;
    {   // 256 threads x one float4 each = full 16x64 tile (staged before aliased agg stores)
        const int i = threadIdx.x;
        const int r = i >> 4, c4 = (i & 15) * 4;
        const float4 v = *(const float4*)(X + (size_t)(rowbase + r) * 64 + c4);
        *(float4*)(xs + r * 68 + c4) = v;
    }
    conv_phase<64, 16, 64, 0>(xs, 68, wl, Wq, bq, Wk, bk, qkv, agg, rowbase);
    conv_phase<64, 16, 64, 1>(xs, 68, wl, Wv, bv, Ws, bs, qkv, agg, rowbase);
}

// ---------------- attention (wave32 per edge) ----------------
__global__ void k_att_scores(const float* qkv, const int* edge, float* scores, float* m) {
    const int e = (blockIdx.x * blockDim.x + threadIdx.x) >> 5;
    const int lane = threadIdx.x & 31;
    if (e >= NE) return;
    const int src = edge[e], dst = edge[NE + e];
    const float2 qv = *(const float2*)(qkv + (size_t)dst * 192 + 2 * lane);
    const float2 kv = *(const float2*)(qkv + (size_t)src * 192 + 64 + 2 * lane);
    float p = qv.x * kv.x + qv.y * kv.y;
    for (int off = 16; off; off >>= 1) p += __shfl_down(p, off, 32);
    if (lane == 0) {
        const float s = p * 0.125f;                    // 1/sqrt(64)
        scores[e] = s;
        atomicMaxFloat(&m[dst], s);
    }
}

__global__ void k_att_exp(const int* edge, float* scores, const float* m, float* denom) {
    const int e = blockIdx.x * blockDim.x + threadIdx.x;
    if (e >= NE) return;
    const int dst = edge[NE + e];
    const float ex = __expf(scores[e] - m[dst]);
    scores[e] = ex;
    atomicAdd(&denom[dst], ex);
}

__global__ void k_att_agg(const float* qkv, const int* edge, const float* scores,
                          const float* denom, float* agg) {
    const int e = (blockIdx.x * blockDim.x + threadIdx.x) >> 5;
    const int lane = threadIdx.x & 31;
    if (e >= NE) return;
    const int src = edge[e], dst = edge[NE + e];
    const float w = scores[e] / (denom[dst] + 1e-16f);
    const float2 vv = *(const float2*)(qkv + (size_t)src * 192 + 128 + 2 * lane);
    atomicAdd(&agg[(size_t)dst * 64 + 2 * lane],     w * vv.x);
    atomicAdd(&agg[(size_t)dst * 64 + 2 * lane + 1], w * vv.y);
}

__global__ void k_silu_ip(float* x, int n) {
    const int i = blockIdx.x * blockDim.x + threadIdx.x;
    if (i < n) x[i] = silu_f(x[i]);
}

__global__ void k_init_md(float* m, float* denom) {
    const int i = blockIdx.x * blockDim.x + threadIdx.x;
    if (i < NN) { m[i] = -INFINITY; denom[i] = 0.f; }
}

// ---------------- pooling ----------------
__global__ void k_init_pool(float* sum, float* sq, float* cnt) {
    const int i = blockIdx.x * blockDim.x + threadIdx.x;
    if (i < NB * 64) { sum[i] = 0.f; sq[i] = 0.f; }
    if (i < NB) cnt[i] = 0.f;
}

__global__ void k_pool(const float* h, const int* batch, float* sum, float* sq, float* cnt) {
    const int n = (blockIdx.x * blockDim.x + threadIdx.x) >> 5;
    const int lane = threadIdx.x & 31;
    if (n >= NN) return;
    const int b = batch[n];
    const float2 hv = *(const float2*)(h + (size_t)n * 64 + 2 * lane);
    atomicAdd(&sum[b * 64 + 2 * lane],     hv.x);
    atomicAdd(&sum[b * 64 + 2 * lane + 1], hv.y);
    atomicAdd(&sq[b * 64 + 2 * lane],      hv.x * hv.x);
    atomicAdd(&sq[b * 64 + 2 * lane + 1],  hv.y * hv.y);
    if (lane == 0) atomicAdd(&cnt[b], 1.f);
}

// ---------------- head MLPs (tiny, B=1024) ----------------
__global__ void k_glob(const float* gf, const float* Wg, const float* bg, float* glob) {
    const int i = blockIdx.x * blockDim.x + threadIdx.x;
    if (i >= NB * 64) return;
    const int row = i >> 6, c = i & 63;
    float acc = bg[c];
    for (int k = 0; k < GFDI; ++k) acc += gf[row * GFDI + k] * Wg[k * 64 + c];
    glob[i] = silu_f(acc);
}

__global__ void k_final(const float* sum, const float* sq, const float* cnt, const float* glob,
                        const float* backend, const float* precision,
                        const float* Wb, const float* bb, float* finl) {
    const int i = blockIdx.x * blockDim.x + threadIdx.x;
    if (i >= NB * 64) return;
    const int row = i >> 6, o = i & 63;
    const float c = fmaxf(cnt[row], 1.f);
    float acc = bb[o];
    for (int k = 0; k < 64; ++k)                       // mean block
        acc += (sum[row * 64 + k] / c) * Wb[k * 64 + o];
    for (int k = 0; k < 64; ++k) {                     // std block
        const float mean = sum[row * 64 + k] / c;
        const float var  = sq[row * 64 + k] / c - mean * mean;
        acc += sqrtf(fmaxf(var, 1e-6f)) * Wb[(64 + k) * 64 + o];
    }
    for (int k = 0; k < 64; ++k)                       // global-mlp block
        acc += glob[row * 64 + k] * Wb[(128 + k) * 64 + o];
    acc += backend[row]   * Wb[192 * 64 + o];
    acc += precision[row] * Wb[193 * 64 + o];
    finl[i] = silu_f(acc);
}

__global__ void k_out(const float* finl, const float* gf,
                      const float* Wth, const float* bth, const float* Wsc,
                      const float* Wrt, const float* brt, float* out) {
    const int i = blockIdx.x * blockDim.x + threadIdx.x;
    if (i < NB * NTHO) {
        const int row = i / NTHO, j = i % NTHO;
        float acc = bth[j];
        for (int k = 0; k < 64; ++k)   acc += finl[row * 64 + k] * Wth[k * NTHO + j];
        for (int k = 0; k < GFDI; ++k) acc += gf[row * GFDI + k] * Wsc[k * NTHO + j];
        out[i] = acc;
    } else if (i < NB * NTHO + NB) {
        const int row = i - NB * NTHO;
        float acc = brt[0];
        for (int k = 0; k < 64; ++k) acc += finl[row * 64 + k] * Wrt[k];
        out[i] = acc;
    }
}

extern "C" void kernel_launch(void* const* d_in, const int* in_sizes, int n_in,
                              void* d_out, int out_size, void* d_ws, size_t ws_size,
                              hipStream_t stream) {
    (void)in_sizes; (void)n_in; (void)out_size; (void)ws_size;
    const int*   gti    = (const int*)d_in[0];
    const int*   arity  = (const int*)d_in[1];
    const int*   isdir  = (const int*)d_in[2];
    const float* idxn   = (const float*)d_in[3];
    const int*   edge   = (const int*)d_in[4];
    const int*   batch  = (const int*)d_in[5];
    const float* gf     = (const float*)d_in[6];
    const float* bkb    = (const float*)d_in[7];
    const float* prb    = (const float*)d_in[8];
    const float* emb    = (const float*)d_in[9];
    const float* W1q = (const float*)d_in[10]; const float* b1q = (const float*)d_in[11];
    const float* W1k = (const float*)d_in[12]; const float* b1k = (const float*)d_in[13];
    const float* W1v = (const float*)d_in[14]; const float* b1v = (const float*)d_in[15];
    const float* W1s = (const float*)d_in[16]; const float* b1s = (const float*)d_in[17];
    const float* W2q = (const float*)d_in[18]; const float* b2q = (const float*)d_in[19];
    const float* W2k = (const float*)d_in[20]; const float* b2k = (const float*)d_in[21];
    const float* W2v = (const float*)d_in[22]; const float* b2v = (const float*)d_in[23];
    const float* W2s = (const float*)d_in[24]; const float* b2s = (const float*)d_in[25];
    const float* Wg  = (const float*)d_in[26]; const float* bg  = (const float*)d_in[27];
    const float* Wb  = (const float*)d_in[28]; const float* bb  = (const float*)d_in[29];
    const float* Wth = (const float*)d_in[30]; const float* bth = (const float*)d_in[31];
    const float* Wsc = (const float*)d_in[32];
    const float* Wrt = (const float*)d_in[33]; const float* brt = (const float*)d_in[34];
    float* out = (float*)d_out;

    // workspace layout (floats)
    float* wsf    = (float*)d_ws;
    float* qkv    = wsf;                           // N*192
    float* agg    = qkv    + (size_t)NN * 192;     // N*64  (also h1, h2)
    float* scores = agg    + (size_t)NN * 64;      // E
    float* mbuf   = scores + (size_t)NE;           // N
    float* dbuf   = mbuf   + (size_t)NN;           // N
    float* psum   = dbuf   + (size_t)NN;           // B*64
    float* psq    = psum   + (size_t)NB * 64;      // B*64
    float* pcnt   = psq    + (size_t)NB * 64;      // B
    float* glob   = pcnt   + (size_t)NB;           // B*64
    float* finl   = glob   + (size_t)NB * 64;      // B*64

    const int T = 256;
    const int gN    = NN / T;            // 1024
    const int gGemm = NN / 16;           // 16384
    const int gEw   = (NE * 32) / T;     // 262144 (wave per edge)
    const int gE    = NE / T;            // 8192
    const int gH    = (NN * 64) / T;     // 65536
    const int gPool = (NN * 32) / T;     // 32768
    const int gB64  = (NB * 64) / T;     // 256

    // ---- layer 1 ----
    k_init_md<<<gN, T, 0, stream>>>(mbuf, dbuf);
    k_gemm_conv1<<<gGemm, T, 0, stream>>>(gti, arity, isdir, idxn, emb,
                                          W1q, b1q, W1k, b1k, W1v, b1v, W1s, b1s, qkv, agg);
    k_att_scores<<<gEw, T, 0, stream>>>(qkv, edge, scores, mbuf);
    k_att_exp<<<gE, T, 0, stream>>>(edge, scores, mbuf, dbuf);
    k_att_agg<<<gEw, T, 0, stream>>>(qkv, edge, scores, dbuf, agg);
    k_silu_ip<<<gH, T, 0, stream>>>(agg, NN * 64);

    // ---- layer 2 (agg doubles as X=h1 and skip-output; safe via LDS staging) ----
    k_init_md<<<gN, T, 0, stream>>>(mbuf, dbuf);
    k_gemm_conv2<<<gGemm, T, 0, stream>>>(agg, W2q, b2q, W2k, b2k, W2v, b2v, W2s, b2s, qkv, agg);
    k_att_scores<<<gEw, T, 0, stream>>>(qkv, edge, scores, mbuf);
    k_att_exp<<<gE, T, 0, stream>>>(edge, scores, mbuf, dbuf);
    k_att_agg<<<gEw, T, 0, stream>>>(qkv, edge, scores, dbuf, agg);
    k_silu_ip<<<gH, T, 0, stream>>>(agg, NN * 64);

    // ---- pooling + head ----
    k_init_pool<<<gB64, T, 0, stream>>>(psum, psq, pcnt);
    k_pool<<<gPool, T, 0, stream>>>(agg, batch, psum, psq, pcnt);
    k_glob<<<gB64, T, 0, stream>>>(gf, Wg, bg, glob);
    k_final<<<gB64, T, 0, stream>>>(psum, psq, pcnt, glob, bkb, prb, Wb, bb, finl);
    k_out<<<(NB * NTHO + NB + T - 1) / T, T, 0, stream>>>(finl, gf, Wth, bth, Wsc, Wrt, brt, out);
}